// HiPPO_LegT_Fast_47536698032542
// MI455X (gfx1250) — compile-verified
//
#include <hip/hip_runtime.h>

// ---------------------------------------------------------------------------
// HiPPO-LegT "scan" recast as a causal-convolution GEMM for MI455X (gfx1250).
//   c_t = sum_{d<=t} (A^d B) * f_{t-d}
// K = [A^d B] (512 x 256) is precomputed with f32 WMMA (repeated squaring +
// block recurrence), then the main kernel is a fully parallel lower-triangular
// Toeplitz GEMM using V_WMMA_F32_16X16X4_F32, f32 end-to-end.
// ---------------------------------------------------------------------------

typedef float v2f __attribute__((ext_vector_type(2)));
typedef float v8f __attribute__((ext_vector_type(8)));

#define L_SZ 512
#define N_ORD 256
#define BC_ROWS 512 // 8 * 64

__device__ __forceinline__ v8f wmma_f32(v2f a, v2f b, v8f c) {
  // D(16x16,f32) = A(16x4,f32) x B(4x16,f32) + C
  return __builtin_amdgcn_wmma_f32_16x16x4_f32(
      /*neg_a=*/false, a, /*neg_b=*/false, b,
      /*c_mod=*/(short)0, c, /*reuse_a=*/false, /*reuse_b=*/false);
}

// ---------------------------------------------------------------------------
// Phase 1a: K_j = A^j B for j = 0..31 (serial, tiny: 32 mat-vecs).
// ---------------------------------------------------------------------------
__global__ __launch_bounds__(256) void k0_kernel(const float* __restrict__ A,
                                                 const float* __restrict__ Bv,
                                                 float* __restrict__ K) {
  __shared__ float kcur[N_ORD];
  const int n = threadIdx.x;
  kcur[n] = Bv[n];
  __syncthreads();
  for (int j = 0; j < 32; ++j) {
    K[j * N_ORD + n] = kcur[n];
    if (j < 31) {
      float s = 0.f;
      for (int i = 0; i < N_ORD; ++i) s = fmaf(A[n * N_ORD + i], kcur[i], s);
      __syncthreads();
      kcur[n] = s;
      __syncthreads();
    }
  }
}

// ---------------------------------------------------------------------------
// Phase 1b: 256x256 f32 GEMM (C = X @ Y) via WMMA. Used for A^2..A^32.
// 16 blocks x 16 waves = 256 waves, one 16x16 tile each, K=256 in 64 chunks.
// ---------------------------------------------------------------------------
__global__ __launch_bounds__(512) void mm256(const float* __restrict__ X,
                                             const float* __restrict__ Y,
                                             float* __restrict__ C) {
  const int lane = threadIdx.x & 31;
  const int wv   = threadIdx.x >> 5;
  const int wg   = blockIdx.x * 16 + wv;
  const int mt = (wg >> 4) * 16;
  const int nt = (wg & 15) * 16;
  const int m  = lane & 15;   // M row (A) or N col (B/D) for this lane
  const int hi = lane >> 4;   // K-pair selector

  v8f acc = {};
  for (int kc = 0; kc < 64; ++kc) {
    const int kb = kc * 4 + 2 * hi;
    // A fragment: lanes 0-15 -> K={0,1}, lanes 16-31 -> K={2,3}
    v2f a = *(const v2f*)(X + (mt + m) * N_ORD + kb);
    // B fragment: same K striping, lane indexes column
    v2f b;
    b.x = Y[kb * N_ORD + nt + m];
    b.y = Y[(kb + 1) * N_ORD + nt + m];
    acc = wmma_f32(a, b, acc);
  }
#pragma unroll
  for (int v = 0; v < 8; ++v)
    C[(mt + v + 8 * hi) * N_ORD + nt + m] = acc[v];
}

// ---------------------------------------------------------------------------
// Phase 1c: K-block recurrence. Kblk_b (256 x 32 cols) = A^32 @ Kblk_{b-1},
// b = 1..15. One workgroup, 32 waves, double-buffered 64KB LDS (CDNA5 WGP
// LDS is 320KB). Wave = (row-tile, col-half).
// ---------------------------------------------------------------------------
__global__ __launch_bounds__(1024) void kblocks_kernel(const float* __restrict__ A32,
                                                       float* __restrict__ K) {
  __shared__ float buf[2][32][N_ORD]; // [colblock j][state n], 64 KB
  for (int idx = threadIdx.x; idx < 32 * N_ORD; idx += 1024)
    buf[0][idx >> 8][idx & 255] = K[idx];
  __syncthreads();

  const int lane = threadIdx.x & 31;
  const int wv   = threadIdx.x >> 5;      // 0..31
  const int r0   = (wv & 15) * 16;        // state-row tile
  const int cb   = (wv >> 4) * 16;        // column half of the 32-col block
  const int m  = lane & 15;
  const int hi = lane >> 4;

  for (int b = 1; b < 16; ++b) {
    const int cur = (b - 1) & 1, nxt = b & 1;
    v8f acc = {};
    for (int kc = 0; kc < 64; ++kc) {
      const int kb = kc * 4 + 2 * hi;
      v2f a = *(const v2f*)(A32 + (r0 + m) * N_ORD + kb);        // L2-hot
      v2f bb = *(const v2f*)(&buf[cur][cb + m][kb]);             // LDS
      acc = wmma_f32(a, bb, acc);
    }
#pragma unroll
    for (int v = 0; v < 8; ++v) {
      const int d   = b * 32 + cb + m;   // K row index (which A^d B)
      const int col = r0 + v + 8 * hi;   // state index n
      K[d * N_ORD + col]    = acc[v];
      buf[nxt][cb + m][col] = acc[v];
    }
    __syncthreads();
  }
}

// ---------------------------------------------------------------------------
// Phase 2: out[t, r, n] = sum_{d<=t} f[r][t-d] * K[d][n].
// Grid (32 t-tiles, 64 r-groups of 8). 16 waves/WG = 16 n-tiles.
// Each wave holds 8 accumulators (one per r) so one K (B-operand) fragment
// feeds 8 WMMAs. Toeplitz A-operand comes from a zero-padded f window in
// LDS; padding realizes the causal mask, and the chunk count 4*(tt+1) skips
// the all-zero upper triangle at tile granularity.
// ---------------------------------------------------------------------------
__global__ __launch_bounds__(512) void hippo_main(const float* __restrict__ inp,
                                                  const float* __restrict__ K,
                                                  float* __restrict__ out) {
  __shared__ float fbuf[8][2 * L_SZ]; // [r][512 zeros | f row], 32 KB
  const int tt = blockIdx.x; // 0..31
  const int rg = blockIdx.y; // 0..63

  for (int idx = threadIdx.x; idx < 8 * L_SZ; idx += 512) {
    const int r8 = idx >> 9, s = idx & 511;
    fbuf[r8][s] = 0.f;
    fbuf[r8][L_SZ + s] = inp[(rg * 8 + r8) * L_SZ + s];
  }
  __syncthreads();

  const int lane = threadIdx.x & 31;
  const int wv   = threadIdx.x >> 5;
  const int n0   = wv * 16;
  const int m  = lane & 15;
  const int hi = lane >> 4;
  const int t0 = tt * 16;

  v8f acc[8];
#pragma unroll
  for (int r8 = 0; r8 < 8; ++r8) acc[r8] = (v8f){};

  const int nchunks = 4 * (tt + 1); // d <= t0+15 only
  for (int kc = 0; kc < nchunks; ++kc) {
    const int db = kc * 4 + 2 * hi; // this lane's first K index in the chunk
    v2f b;
    b.x = K[db * N_ORD + n0 + m];
    b.y = K[(db + 1) * N_ORD + n0 + m];
    const int j1 = t0 + m - db; // f index for the .x element (may be < 0)
#pragma unroll
    for (int r8 = 0; r8 < 8; ++r8) {
      v2f a;
      a.x = fbuf[r8][L_SZ + j1];     // f[t - db]   (0 if t < db)
      a.y = fbuf[r8][L_SZ + j1 - 1]; // f[t - db-1]
      acc[r8] = wmma_f32(a, b, acc[r8]);
    }
  }

#pragma unroll
  for (int r8 = 0; r8 < 8; ++r8) {
    const long r = rg * 8 + r8;
#pragma unroll
    for (int v = 0; v < 8; ++v) {
      const long t = t0 + v + 8 * hi;
      out[t * (long)(BC_ROWS * N_ORD) + r * N_ORD + n0 + m] = acc[r8][v];
    }
  }
}

// ---------------------------------------------------------------------------
// Launch. d_in = {inputs (8*64*512 f32), A (256*256 f32), B (256 f32)}.
// Workspace: K (512KB) | P1 (256KB) | P2 (256KB)  => 1 MB.
// ---------------------------------------------------------------------------
extern "C" void kernel_launch(void* const* d_in, const int* in_sizes, int n_in,
                              void* d_out, int out_size, void* d_ws, size_t ws_size,
                              hipStream_t stream) {
  (void)in_sizes; (void)n_in; (void)out_size; (void)ws_size;
  const float* inp = (const float*)d_in[0];
  const float* A   = (const float*)d_in[1];
  const float* Bv  = (const float*)d_in[2];
  float* out  = (float*)d_out;
  float* Kker = (float*)d_ws;                    // 512 x 256
  float* P1   = Kker + (size_t)L_SZ * N_ORD;     // 256 x 256
  float* P2   = P1 + (size_t)N_ORD * N_ORD;      // 256 x 256

  k0_kernel<<<1, 256, 0, stream>>>(A, Bv, Kker);     // K_0..K_31
  mm256<<<16, 512, 0, stream>>>(A,  A,  P1);         // A^2
  mm256<<<16, 512, 0, stream>>>(P1, P1, P2);         // A^4
  mm256<<<16, 512, 0, stream>>>(P2, P2, P1);         // A^8
  mm256<<<16, 512, 0, stream>>>(P1, P1, P2);         // A^16
  mm256<<<16, 512, 0, stream>>>(P2, P2, P1);         // A^32
  kblocks_kernel<<<1, 1024, 0, stream>>>(P1, Kker);  // K_32..K_511
  dim3 grid(32, 64);
  hippo_main<<<grid, 512, 0, stream>>>(inp, Kker, out);
}